// RandLANet_38379827757525
// MI455X (gfx1250) — compile-verified
//
#include <hip/hip_runtime.h>
#include <hip/hip_bf16.h>
#include <math.h>
#include <stdint.h>

// ---------------------------------------------------------------------------
// Problem constants (fixed by the reference)
// ---------------------------------------------------------------------------
#define NPTS   20000
#define MQ     5000
#define DX     64
#define KNN    16

typedef __attribute__((ext_vector_type(2))) float v2f;
typedef __attribute__((ext_vector_type(8))) float v8f;
typedef int v4i __attribute__((vector_size(16)));   // matches builtin param type

// ---------------------------------------------------------------------------
// CDNA5 async global->LDS copy (ASYNCcnt-tracked), with portable fallback.
// Signature (from hipcc diagnostic): (AS1 v4i* src, AS3 v4i* dst, imm, imm).
// ---------------------------------------------------------------------------
__device__ __forceinline__ void async_g2l_b128(void* lds_dst, const void* gsrc) {
#if __has_builtin(__builtin_amdgcn_global_load_async_to_lds_b128)
  // AS1/AS3 pointers rebuilt from raw addresses (low 32 bits of a generic LDS
  // pointer are the LDS byte offset per the flat-aperture mapping).
  __attribute__((address_space(1))) v4i* g =
      (__attribute__((address_space(1))) v4i*)(uintptr_t)gsrc;
  __attribute__((address_space(3))) v4i* l =
      (__attribute__((address_space(3))) v4i*)(uint32_t)(uintptr_t)lds_dst;
  __builtin_amdgcn_global_load_async_to_lds_b128(g, l, 0, 0);
#else
  *(float4*)lds_dst = *(const float4*)gsrc;
#endif
}

__device__ __forceinline__ void async_wait0() {
#if __has_builtin(__builtin_amdgcn_s_wait_asynccnt)
  __builtin_amdgcn_s_wait_asynccnt(0);
#else
#if __has_builtin(__builtin_amdgcn_global_load_async_to_lds_b128)
  asm volatile("s_wait_asynccnt 0x0" ::: "memory");
#endif
#endif
}

// ---------------------------------------------------------------------------
// WMMA helpers: V_WMMA_F32_16X16X4_F32  (fp32 matrix core path on gfx1250)
//   A: 16x4  -> 2 VGPR/lane : lanes 0-15 hold K={0,1}, lanes 16-31 K={2,3}
//   B: 4x16  -> 2 VGPR/lane : mirrored (col = lane&15)
//   C/D: 16x16 -> 8 VGPR    : M = r + 8*(lane>=16), N = lane&15
// ---------------------------------------------------------------------------
__device__ __forceinline__ v8f wmma_f32(v2f a, v2f b, v8f c) {
  return __builtin_amdgcn_wmma_f32_16x16x4_f32(
      /*neg_a=*/false, a, /*neg_b=*/false, b,
      /*c_mod=*/(short)0, c, /*reuse_a=*/false, /*reuse_b=*/false);
}

// A fragment from LDS, row-major [16 x lda]
__device__ __forceinline__ v2f lds_load_a(const float* A, int lda, int k0, int lane) {
  int row = lane & 15;
  int kb  = (lane >> 4) << 1;
  const float* p = A + row * lda + k0 + kb;
  v2f a; a.x = p[0]; a.y = p[1];
  return a;
}

// B fragment, row-major [Kdim x ldb] (global or LDS), tile col offset n0
__device__ __forceinline__ v2f load_b(const float* B, int ldb, int k0, int n0, int lane) {
  int col = (lane & 15) + n0;
  int k   = k0 + ((lane >> 4) << 1);
  v2f b; b.x = B[k * ldb + col]; b.y = B[(k + 1) * ldb + col];
  return b;
}

// D tile -> LDS with bias + ReLU.  Dest col offset dn0 (ld = ldd), bias index bn0.
__device__ __forceinline__ void store_tile_relu(float* D, int ldd, int dn0,
                                                const float* bias, int bn0,
                                                v8f acc, int lane) {
  int col = lane & 15;
  int rb  = (lane >> 4) << 3;
  float bv = bias[bn0 + col];
#pragma unroll
  for (int r = 0; r < 8; ++r) {
    float v = acc[r] + bv;
    D[(rb + r) * ldd + dn0 + col] = v > 0.0f ? v : 0.0f;
  }
}

// ---------------------------------------------------------------------------
// Kernel 1: exact KNN (K=16) by brute-force scan with LDS tiling.
// One thread per query; candidate tiles staged through LDS with async
// global->LDS copies. Top-16 kept in a fully unrolled register insertion
// sort (ascending d2), matching jax.lax.top_k(-d2) ordering.
// Also emits pos_q and batch[idx] outputs.
// ---------------------------------------------------------------------------
#define KNN_BLK  128
#define KNN_TILE 512

__global__ __launch_bounds__(KNN_BLK) void knn_kernel(
    const float* __restrict__ pos, const int* __restrict__ idx,
    const int* __restrict__ batch, int* __restrict__ nbr,
    float* __restrict__ pos_q_out, float* __restrict__ batch_out) {
  __shared__ float sp[KNN_TILE * 3];

  const int m   = blockIdx.x * KNN_BLK + threadIdx.x;
  const bool on = (m < MQ);

  float qx = 0.f, qy = 0.f, qz = 0.f;
  int q = 0;
  if (on) {
    q  = idx[m];
    qx = pos[q * 3 + 0];
    qy = pos[q * 3 + 1];
    qz = pos[q * 3 + 2];
  }

  float bd[KNN];
  int   bi[KNN];
#pragma unroll
  for (int i = 0; i < KNN; ++i) { bd[i] = 3.4e38f; bi[i] = 0; }

  for (int t0 = 0; t0 < NPTS; t0 += KNN_TILE) {
    const int npts_tile = min(KNN_TILE, NPTS - t0);
    const int nf4 = (npts_tile * 3) >> 2;   // 384 full tile, 24 for the tail
    // cooperative async tile load: up to 384 float4s / 128 threads = 3 each
#pragma unroll
    for (int i = 0; i < (KNN_TILE * 3) / (4 * KNN_BLK); ++i) {
      int e4 = i * KNN_BLK + threadIdx.x;
      if (e4 < nf4)
        async_g2l_b128(&sp[e4 * 4], &pos[(size_t)t0 * 3 + e4 * 4]);
    }
    // prefetch next tile into the hierarchy (global_prefetch_b8)
    if (t0 + KNN_TILE < NPTS)
      __builtin_prefetch(&pos[(size_t)(t0 + KNN_TILE) * 3 + threadIdx.x], 0, 1);
    async_wait0();
    __syncthreads();

    if (on) {
      for (int c = 0; c < npts_tile; ++c) {
        float dx = qx - sp[c * 3 + 0];
        float dy = qy - sp[c * 3 + 1];
        float dz = qz - sp[c * 3 + 2];
        float d2 = fmaf(dx, dx, fmaf(dy, dy, dz * dz));
        if (d2 < bd[KNN - 1]) {
          bd[KNN - 1] = d2; bi[KNN - 1] = t0 + c;
#pragma unroll
          for (int j = KNN - 1; j > 0; --j) {
            if (bd[j] < bd[j - 1]) {
              float td = bd[j]; bd[j] = bd[j - 1]; bd[j - 1] = td;
              int   ti = bi[j]; bi[j] = bi[j - 1]; bi[j - 1] = ti;
            }
          }
        }
      }
    }
    __syncthreads();
  }

  if (on) {
#pragma unroll
    for (int i = 0; i < KNN; ++i) nbr[m * KNN + i] = bi[i];
    pos_q_out[m * 3 + 0] = qx;
    pos_q_out[m * 3 + 1] = qy;
    pos_q_out[m * 3 + 2] = qz;
    ((int*)batch_out)[m] = batch[q];   // int bit-store into float-typed region
  }
}

// ---------------------------------------------------------------------------
// Kernel 2: per-center edge pipeline. One wave32 per query center m:
//   rel[16x10] --WMMA--> r[16x64] --WMMA--> rij[16x32]
//   fij = [x_j | rij] (16x96) --WMMA--> g[16x96]
//   softmax over features, msg = s*fij, mean over the 16 rows -> aggr[m,96]
// A operands staged in per-wave LDS. Wp1 staged zero-padded (10->12 rows) in
// LDS per block so the K-pad is data, not control flow; Wp2/Wa fragments read
// from global (weights are L2/WGP$-resident). x_j gathered to LDS with async
// copies overlapped with GEMM2. 4 waves/block -> 51KB LDS/block.
// Grid is exactly M/4 blocks, so EXEC is all-ones at every WMMA.
// ---------------------------------------------------------------------------
#define EW 4   // waves per block

__global__ __launch_bounds__(EW * 32) void edge_kernel(
    const float* __restrict__ x,   const float* __restrict__ pos,
    const int*   __restrict__ idx, const int*   __restrict__ nbr,
    const float* __restrict__ Wp1, const float* __restrict__ bp1,
    const float* __restrict__ Wp2, const float* __restrict__ bp2,
    const float* __restrict__ Wa,  const float* __restrict__ ba,
    float* __restrict__ aggr) {
  __shared__ float bufA[EW][16 * 96];
  __shared__ float bufB[EW][16 * 96];
  __shared__ float sWp1[12 * 64];          // zero-padded [12 x 64]

  const int tid  = threadIdx.x;
  const int w    = tid >> 5;
  const int lane = tid & 31;
  const int m    = blockIdx.x * EW + w;     // grid sized so m < MQ always

  float* A  = bufA[w];
  float* Bl = bufB[w];

  // ---- stage Wp1 (zero-padded rows 10,11) : 768 floats / 128 threads
#pragma unroll
  for (int i = 0; i < (12 * 64) / (EW * 32); ++i) {
    int e = i * (EW * 32) + tid;
    sWp1[e] = (e < 10 * 64) ? Wp1[e] : 0.0f;
  }

  // ---- phase 1: rel features [16 x 12] (K padded 10->12), row per neighbor
  if (lane < KNN) {
    int j  = nbr[m * KNN + lane];
    int q  = idx[m];
    float qx = pos[q * 3], qy = pos[q * 3 + 1], qz = pos[q * 3 + 2];
    float px = pos[j * 3], py = pos[j * 3 + 1], pz = pos[j * 3 + 2];
    float vx = qx - px, vy = qy - py, vz = qz - pz;
    float d  = sqrtf(fmaf(vx, vx, fmaf(vy, vy, vz * vz)));
    float* rr = &A[lane * 12];
    rr[0] = qx; rr[1] = qy; rr[2] = qz;
    rr[3] = px; rr[4] = py; rr[5] = pz;
    rr[6] = vx; rr[7] = vy; rr[8] = vz;
    rr[9] = d;  rr[10] = 0.f; rr[11] = 0.f;
  }
  __syncthreads();

  // ---- GEMM1: rel[16x12] @ Wp1[12x64] -> r[16x64] (relu) into bufB
#pragma unroll
  for (int nt = 0; nt < 4; ++nt) {
    v8f acc = {0.f, 0.f, 0.f, 0.f, 0.f, 0.f, 0.f, 0.f};
#pragma unroll
    for (int k0 = 0; k0 < 12; k0 += 4) {
      v2f a = lds_load_a(A, 12, k0, lane);
      v2f b = load_b(sWp1, 64, k0, nt * 16, lane);
      acc = wmma_f32(a, b, acc);
    }
    store_tile_relu(Bl, 64, nt * 16, bp1, nt * 16, acc, lane);
  }
  __syncthreads();

  // ---- async-gather x_j into fij cols [0,64) of bufA (rel is dead), ld = 96
  {
    int row = lane & 15;
    int c0  = (lane >> 4) * 32;
    int j   = nbr[m * KNN + row];
#pragma unroll
    for (int i = 0; i < 8; ++i)
      async_g2l_b128(&A[row * 96 + c0 + i * 4],
                     &x[(size_t)j * DX + c0 + i * 4]);
  }

  // ---- GEMM2 (overlaps the async gather):
  //      r[16x64] @ Wp2[64x32] -> rij -> fij cols [64,96) (relu)
#pragma unroll
  for (int nt = 0; nt < 2; ++nt) {
    v8f acc = {0.f, 0.f, 0.f, 0.f, 0.f, 0.f, 0.f, 0.f};
#pragma unroll
    for (int k0 = 0; k0 < 64; k0 += 4) {
      v2f a = lds_load_a(Bl, 64, k0, lane);
      v2f b = load_b(Wp2, 32, k0, nt * 16, lane);
      acc = wmma_f32(a, b, acc);
    }
    store_tile_relu(A, 96, 64 + nt * 16, bp2, nt * 16, acc, lane);
  }
  async_wait0();
  __syncthreads();

  // ---- GEMM3: fij[16x96] @ Wa[96x96] -> g[16x96] (relu) into bufB
#pragma unroll
  for (int nt = 0; nt < 6; ++nt) {
    v8f acc = {0.f, 0.f, 0.f, 0.f, 0.f, 0.f, 0.f, 0.f};
#pragma unroll
    for (int k0 = 0; k0 < 96; k0 += 4) {
      v2f a = lds_load_a(A, 96, k0, lane);
      v2f b = load_b(Wa, 96, k0, nt * 16, lane);
      acc = wmma_f32(a, b, acc);
    }
    store_tile_relu(Bl, 96, nt * 16, ba, nt * 16, acc, lane);
  }
  __syncthreads();

  // ---- softmax over 96 features per row, msg = s * fij (in-place in bufB)
  //      2 lanes per row: lane pair (r, r+16) each handles 48 columns.
  {
    int row = lane & 15;
    int cb  = (lane >> 4) * 48;
    float*       gr = &Bl[row * 96];
    const float* fr = &A[row * 96];
    float mx = -3.4e38f;
    for (int c = 0; c < 48; ++c) mx = fmaxf(mx, gr[cb + c]);
    mx = fmaxf(mx, __shfl_xor(mx, 16, 32));
    float s = 0.f;
    for (int c = 0; c < 48; ++c) {
      float e = __expf(gr[cb + c] - mx);
      gr[cb + c] = e;
      s += e;
    }
    s += __shfl_xor(s, 16, 32);
    float inv = 1.0f / s;
    for (int c = 0; c < 48; ++c) gr[cb + c] = gr[cb + c] * inv * fr[cb + c];
  }
  __syncthreads();

  // ---- mean over the 16 neighbor rows -> aggr[m, 96] (3 cols per lane)
#pragma unroll
  for (int t = 0; t < 3; ++t) {
    int col = lane * 3 + t;
    float s = 0.f;
#pragma unroll
    for (int r = 0; r < KNN; ++r) s += Bl[r * 96 + col];
    aggr[m * 96 + col] = s * (1.0f / (float)KNN);
  }
}

// ---------------------------------------------------------------------------
// Kernel 3: out = relu(aggr[5000x96] @ Wg[96x128] + bg).
// One 16x16 WMMA tile per wave iteration; 4 waves cover the 8 col-tiles of a
// 16-row stripe in 2 uniform iterations (EXEC stays all-ones; row bounds are
// handled with selects/predicated stores only).
// ---------------------------------------------------------------------------
__global__ __launch_bounds__(128) void out_kernel(
    const float* __restrict__ aggr, const float* __restrict__ Wg,
    const float* __restrict__ bg, float* __restrict__ out) {
  const int tid  = threadIdx.x;
  const int w    = tid >> 5;
  const int lane = tid & 31;
  const int row0 = blockIdx.x * 16;

  for (int ct = w; ct < 8; ct += 4) {        // exactly 2 uniform iterations/wave
    const int n0 = ct * 16;
    v8f acc = {0.f, 0.f, 0.f, 0.f, 0.f, 0.f, 0.f, 0.f};
    const int ar = row0 + (lane & 15);
    const int kb = (lane >> 4) << 1;
#pragma unroll
    for (int k0 = 0; k0 < 96; k0 += 4) {
      v2f a;
      a.x = (ar < MQ) ? aggr[ar * 96 + k0 + kb]     : 0.0f;
      a.y = (ar < MQ) ? aggr[ar * 96 + k0 + kb + 1] : 0.0f;
      v2f b = load_b(Wg, 128, k0, n0, lane);
      acc = wmma_f32(a, b, acc);
    }
    const int col = lane & 15;
    const int rb  = (lane >> 4) << 3;
    const float bv = bg[n0 + col];
#pragma unroll
    for (int r = 0; r < 8; ++r) {
      int rr = row0 + rb + r;
      if (rr < MQ) {
        float v = acc[r] + bv;
        out[rr * 128 + n0 + col] = v > 0.0f ? v : 0.0f;
      }
    }
  }
}

// ---------------------------------------------------------------------------
// Launcher
// ---------------------------------------------------------------------------
extern "C" void kernel_launch(void* const* d_in, const int* in_sizes, int n_in,
                              void* d_out, int out_size, void* d_ws, size_t ws_size,
                              hipStream_t stream) {
  const float* x     = (const float*)d_in[0];
  const float* pos   = (const float*)d_in[1];
  const int*   batch = (const int*)  d_in[2];
  const int*   idx   = (const int*)  d_in[3];
  const float* Wp1   = (const float*)d_in[4];
  const float* bp1   = (const float*)d_in[5];
  const float* Wp2   = (const float*)d_in[6];
  const float* bp2   = (const float*)d_in[7];
  const float* Wa    = (const float*)d_in[8];
  const float* ba    = (const float*)d_in[9];
  const float* Wg    = (const float*)d_in[10];
  const float* bg    = (const float*)d_in[11];

  float* out     = (float*)d_out;                 // [MQ,128]
  float* pos_q_o = out + (size_t)MQ * 128;        // [MQ,3]
  float* batch_o = pos_q_o + (size_t)MQ * 3;      // [MQ] (int bits)

  float* aggr = (float*)d_ws;                     // [MQ,96]
  int*   nbr  = (int*)((char*)d_ws + (size_t)MQ * 96 * sizeof(float)); // [MQ,16]

  knn_kernel<<<(MQ + KNN_BLK - 1) / KNN_BLK, KNN_BLK, 0, stream>>>(
      pos, idx, batch, nbr, pos_q_o, batch_o);

  edge_kernel<<<MQ / EW, EW * 32, 0, stream>>>(
      x, pos, idx, nbr, Wp1, bp1, Wp2, bp2, Wa, ba, aggr);

  out_kernel<<<(MQ + 15) / 16, 128, 0, stream>>>(aggr, Wg, bg, out);
}